// FilterDetections_46729244181053
// MI455X (gfx1250) — compile-verified
//
#include <hip/hip_runtime.h>
#include <stdint.h>

#define N_ANCHORS 50000
#define N_CLASSES 80
#define MAX_DET   100
#define SCORE_THR 0.05f
#define BLK       256
#define CHUNK     2048   // anchors staged per TDM transfer
#define CAND_CAP  4096   // power of two, LDS candidate buffer
#define KEEP      1024   // survivors after a flush
#define FLUSH_AT  (CAND_CAP - CHUNK)   // flush before overflow possible

#if defined(__has_builtin)
#if __has_builtin(__builtin_amdgcn_tensor_load_to_lds)
#define HAVE_TDM 1
#endif
#endif

typedef unsigned int uint32x4 __attribute__((ext_vector_type(4)));
typedef int          int32x8  __attribute__((ext_vector_type(8)));
typedef int          int32x4  __attribute__((ext_vector_type(4)));

// ---- order-preserving float <-> uint key helpers -------------------------
// enc: larger score -> smaller key (descending); ties broken by low 32 bits
// (anchor index / flat position, ascending) = argmax/top_k first-occurrence.
__device__ __forceinline__ unsigned enc_desc(float f) {
  unsigned u = __float_as_uint(f);
  unsigned o = (u & 0x80000000u) ? ~u : (u | 0x80000000u);
  return ~o;
}
__device__ __forceinline__ float dec_desc(unsigned k) {
  unsigned o = ~k;
  unsigned u = (o & 0x80000000u) ? (o & 0x7fffffffu) : ~o;
  return __uint_as_float(u);
}

#define PADKEY 0xffffffffffffffffull

// ---- in-LDS bitonic sort (ascending on u64 keys), BLK threads ------------
template <int NSORT>
__device__ void bitonic_sort(unsigned long long* a) {
  for (unsigned k = 2; k <= (unsigned)NSORT; k <<= 1) {
    for (unsigned j = k >> 1; j > 0; j >>= 1) {
      for (unsigned i = threadIdx.x; i < (unsigned)NSORT; i += BLK) {
        unsigned ixj = i ^ j;
        if (ixj > i) {
          unsigned long long x = a[i], y = a[ixj];
          bool up = ((i & k) == 0);
          if ((x > y) == up) { a[i] = y; a[ixj] = x; }
        }
      }
      __syncthreads();
    }
  }
}

// ---- staging: strided class-column chunk -> LDS --------------------------
#ifdef HAVE_TDM
// TDM descriptor gather: tile_dim0 = 1 element, tile_dim1 = CHUNK rows,
// tensor_dim0_stride = N_CLASSES  => 4B elements every 320B, packed into LDS.
// tensor_dim1 = rows remaining => hardware zero-fills past the tensor end.
__device__ __forceinline__ void stage_issue(const float* gp, unsigned lds_off,
                                            int rows_rem) {
  unsigned long long ga = (unsigned long long)(uintptr_t)gp;
  uint32x4 g0;
  g0[0] = 1u;                                          // count=1, valid user D#
  g0[1] = lds_off;                                     // lds_addr (bytes)
  g0[2] = (unsigned)(ga & 0xffffffffu);                // global_addr[31:0]
  g0[3] = (unsigned)((ga >> 32) & 0x01ffffffu)         // global_addr[56:32]
        | (2u << 30);                                  // type = 2 ("image")
  int32x8 g1;
  g1[0] = (int)(2u << 16);                             // data_size = 4 bytes
  g1[1] = (int)((unsigned)N_CLASSES << 16);            // tensor_dim0 lo16 = 80
  g1[2] = (int)(((unsigned)rows_rem & 0xffffu) << 16); // tensor_dim1 lo16
  g1[3] = (int)((((unsigned)rows_rem >> 16) & 0xffffu) // tensor_dim1 hi16
        | (1u << 16));                                 // tile_dim0 = 1
  g1[4] = CHUNK;                                       // tile_dim1 (tile_dim2=0)
  g1[5] = N_CLASSES;                                   // dim0 stride lo32 = 80
  g1[6] = 0;                                           // stride hi / dim1 stride
  g1[7] = 0;
  int32x4 z4 = {0, 0, 0, 0};
  int32x8 z8 = {0, 0, 0, 0, 0, 0, 0, 0};
  // amdgpu-toolchain (clang-23) 6-arg form: (g0, g1, g2, g3, extra, cpol)
  __builtin_amdgcn_tensor_load_to_lds(g0, g1, z4, z4, z8, 0);
}
__device__ __forceinline__ void stage_wait() {
#if __has_builtin(__builtin_amdgcn_s_wait_tensorcnt)
  __builtin_amdgcn_s_wait_tensorcnt(0);
#else
  asm volatile("s_wait_tensorcnt 0" ::: "memory");
#endif
}
#endif

// ==========================================================================
// Kernel 1: per (batch, class) streaming candidate selection + greedy NMS.
// grid = (C, B), block = 256.
// ==========================================================================
__global__ __launch_bounds__(BLK) void nms_per_class(
    const float* __restrict__ boxes,        // [B, N, 4]
    const float* __restrict__ cls,          // [B, N, C]
    float* __restrict__ det_score,          // [B, C, MAX_DET]
    int*   __restrict__ det_idx)            // [B, C, MAX_DET]
{
  __shared__ unsigned long long cand[CAND_CAP];
  __shared__ float stage[2][CHUNK];
  __shared__ int   count;
  __shared__ float thresh;
  __shared__ float sx1[MAX_DET], sy1[MAX_DET], sx2[MAX_DET], sy2[MAX_DET];
  __shared__ float osc[MAX_DET];
  __shared__ int   oid[MAX_DET];

  const int c   = blockIdx.x;
  const int b   = blockIdx.y;
  const int tid = threadIdx.x;
  const float* clsb = cls + ((size_t)b * N_ANCHORS) * N_CLASSES + c;
  const float* boxb = boxes + (size_t)b * N_ANCHORS * 4;

  for (int i = tid; i < CAND_CAP; i += BLK) cand[i] = PADKEY;
  if (tid == 0) { count = 0; thresh = SCORE_THR; }

  // ---- prologue: stage chunk 0 -------------------------------------------
#ifdef HAVE_TDM
  if (tid < 32) {
    stage_issue(clsb, (unsigned)(uintptr_t)(&stage[0][0]), N_ANCHORS);
    stage_wait();
  }
#else
  for (int r = tid; r < CHUNK; r += BLK) {
    int n = (r < N_ANCHORS) ? r : (N_ANCHORS - 1);
    unsigned lds_off = (unsigned)(uintptr_t)(&stage[0][r]);
    unsigned long long ga = (unsigned long long)(uintptr_t)(clsb + (size_t)n * N_CLASSES);
    asm volatile("global_load_async_to_lds_b32 %0, %1, off"
                 :: "v"(lds_off), "v"(ga) : "memory");
  }
  asm volatile("s_wait_asynccnt 0" ::: "memory");
#endif
  __syncthreads();

  // ---- streaming pass with adaptive threshold ----------------------------
  int buf = 0;
  for (int base = 0; base < N_ANCHORS; base += CHUNK) {
    const int nbase = base + CHUNK;
    if (nbase < N_ANCHORS) {
      // kick off DMA of the next chunk; overlaps filter + flush below
#ifdef HAVE_TDM
      if (tid < 32)
        stage_issue(clsb + (size_t)nbase * N_CLASSES,
                    (unsigned)(uintptr_t)(&stage[buf ^ 1][0]),
                    N_ANCHORS - nbase);
#else
      for (int r = tid; r < CHUNK; r += BLK) {
        int n = nbase + r;
        if (n >= N_ANCHORS) n = N_ANCHORS - 1;
        unsigned lds_off = (unsigned)(uintptr_t)(&stage[buf ^ 1][r]);
        unsigned long long ga = (unsigned long long)(uintptr_t)(clsb + (size_t)n * N_CLASSES);
        asm volatile("global_load_async_to_lds_b32 %0, %1, off"
                     :: "v"(lds_off), "v"(ga) : "memory");
      }
#endif
    }

    const float th = thresh;
    for (int r = tid; r < CHUNK; r += BLK) {
      const int n = base + r;
      const float s = stage[buf][r];
      if (n < N_ANCHORS && s > th) {
        int p = atomicAdd(&count, 1);
        cand[p] = (((unsigned long long)enc_desc(s)) << 32) | (unsigned)n;
      }
    }
    __syncthreads();

    if (count > FLUSH_AT) {
      bitonic_sort<CAND_CAP>(cand);                 // best (smallest key) first
      if (tid == 0) {
        thresh = dec_desc((unsigned)(cand[KEEP - 1] >> 32));
        count  = KEEP;
      }
      for (int i = KEEP + tid; i < CAND_CAP; i += BLK) cand[i] = PADKEY;
      __syncthreads();
    }

    // make the next staged buffer visible before anyone reads it
#ifdef HAVE_TDM
    if (tid < 32) stage_wait();
#else
    asm volatile("s_wait_asynccnt 0" ::: "memory");
#endif
    __syncthreads();
    buf ^= 1;
  }
  bitonic_sort<CAND_CAP>(cand);   // final: score desc, index asc

  // ---- sorted greedy NMS, wave 0 only ------------------------------------
  if (tid < 32) {
    const int lane = tid;
    int nsel = 0;
    for (int i = 0; i < CAND_CAP && nsel < MAX_DET; ++i) {
      unsigned long long k = cand[i];
      if (k == PADKEY) break;
      float s = dec_desc((unsigned)(k >> 32));
      int idx = (int)(k & 0xffffffffu);
      float bx1 = boxb[idx * 4 + 0], by1 = boxb[idx * 4 + 1];
      float bx2 = boxb[idx * 4 + 2], by2 = boxb[idx * 4 + 3];
      float barea = (bx2 - bx1) * (by2 - by1);
      bool sup = false;
      for (int j = lane; j < nsel; j += 32) {
        float ix1 = fmaxf(bx1, sx1[j]);
        float iy1 = fmaxf(by1, sy1[j]);
        float ix2 = fminf(bx2, sx2[j]);
        float iy2 = fminf(by2, sy2[j]);
        float inter = fmaxf(ix2 - ix1, 0.0f) * fmaxf(iy2 - iy1, 0.0f);
        float sarea = (sx2[j] - sx1[j]) * (sy2[j] - sy1[j]);
        float uni = fmaxf(barea + sarea - inter, 1e-8f);
        if (inter > 0.5f * uni) sup = true;   // == (iou > NMS_THRESHOLD)
      }
      if (__ballot(sup) == 0ull) {
        // every lane writes identical values -> each lane's later reads are
        // satisfied by its own program-order LDS writes
        sx1[nsel] = bx1; sy1[nsel] = by1; sx2[nsel] = bx2; sy2[nsel] = by2;
        osc[nsel] = s;   oid[nsel] = idx;
        ++nsel;
      }
    }
    for (int m = nsel + lane; m < MAX_DET; m += 32) {
      osc[m] = -INFINITY;     // reference pads with -inf / dummy anchor
      oid[m] = N_ANCHORS;
    }
  }
  __syncthreads();

  const size_t obase = ((size_t)b * N_CLASSES + c) * MAX_DET;
  for (int m = tid; m < MAX_DET; m += BLK) {
    det_score[obase + m] = osc[m];
    det_idx[obase + m]   = oid[m];
  }
}

// ==========================================================================
// Kernel 2: per batch cross-class top-100 + gather. grid = (B), block = 256.
// ==========================================================================
#define TOPN 8192
__global__ __launch_bounds__(BLK) void topk_and_gather(
    const float* __restrict__ boxes,       // [B, N, 4]
    const float* __restrict__ det_score,   // [B, C*MAX_DET]
    const int*   __restrict__ det_idx,     // [B, C*MAX_DET]
    float* __restrict__ out_boxes,         // [B, MAX_DET, 4]
    float* __restrict__ out_scores,        // [B, MAX_DET]
    int*   __restrict__ out_labels)        // [B, MAX_DET]
{
  __shared__ unsigned long long key[TOPN];
  const int b = blockIdx.x, tid = threadIdx.x;
  const int M = N_CLASSES * MAX_DET;       // 8000
  const float* ds = det_score + (size_t)b * M;
  const int*   di = det_idx   + (size_t)b * M;

  for (int e = tid; e < TOPN; e += BLK) {
    if (e < M) {
      float s = ds[e];
      key[e] = (((unsigned long long)enc_desc(s)) << 32) | (unsigned)e;
    } else {
      key[e] = PADKEY;
    }
  }
  __syncthreads();
  bitonic_sort<TOPN>(key);   // score desc, flat position asc (lax.top_k ties)

  if (tid < MAX_DET) {
    unsigned long long k = key[tid];
    int pos   = (int)(k & 0xffffffffu);
    float s   = dec_desc((unsigned)(k >> 32));
    int c     = pos / MAX_DET;
    int idx   = di[pos];
    int label = (idx < N_ANCHORS) ? c : -1;
    float bx1 = 0.f, by1 = 0.f, bx2 = 0.f, by2 = 0.f;   // boxes_c dummy row
    if (idx < N_ANCHORS) {
      const float* bp = boxes + ((size_t)b * N_ANCHORS + idx) * 4;
      bx1 = bp[0]; by1 = bp[1]; bx2 = bp[2]; by2 = bp[3];
    }
    const int o = b * MAX_DET + tid;
    out_boxes[o * 4 + 0] = bx1;
    out_boxes[o * 4 + 1] = by1;
    out_boxes[o * 4 + 2] = bx2;
    out_boxes[o * 4 + 3] = by2;
    out_scores[o] = s;
    out_labels[o] = label;
  }
}

// ==========================================================================
extern "C" void kernel_launch(void* const* d_in, const int* in_sizes, int n_in,
                              void* d_out, int out_size, void* d_ws, size_t ws_size,
                              hipStream_t stream) {
  const float* boxes = (const float*)d_in[0];          // [B, N, 4] f32
  const float* cls   = (const float*)d_in[1];          // [B, N, C] f32
  const int B = in_sizes[1] / (N_ANCHORS * N_CLASSES);

  // workspace layout: det_score [B*C*100] f32 | det_idx [B*C*100] i32
  float* det_score = (float*)d_ws;
  int*   det_idx   = (int*)((char*)d_ws +
                     (size_t)B * N_CLASSES * MAX_DET * sizeof(float));

  // output layout (tuple concatenated flat): boxes | scores | labels
  float* out_boxes  = (float*)d_out;
  float* out_scores = out_boxes + (size_t)B * MAX_DET * 4;
  int*   out_labels = (int*)(out_scores + (size_t)B * MAX_DET);

  dim3 g1(N_CLASSES, B);
  nms_per_class<<<g1, dim3(BLK), 0, stream>>>(boxes, cls, det_score, det_idx);
  topk_and_gather<<<dim3(B), dim3(BLK), 0, stream>>>(boxes, det_score, det_idx,
                                                     out_boxes, out_scores, out_labels);
}